// HyperAggregator_32117765440056
// MI455X (gfx1250) — compile-verified
//
#include <hip/hip_runtime.h>

#define N_ENT 100000
#define N_HE  20000
#define DIM   128

typedef float v2f __attribute__((ext_vector_type(2)));
typedef float v8f __attribute__((ext_vector_type(8)));

// ---------------------------------------------------------------------------
// Zero the workspace accumulators (side, he1, he2). Grid-stride float4 stores.
// ---------------------------------------------------------------------------
__global__ void zero_ws_kernel(float4* __restrict__ ws, long n4) {
    long i = (long)blockIdx.x * blockDim.x + threadIdx.x;
    long stride = (long)gridDim.x * blockDim.x;
    float4 z = make_float4(0.f, 0.f, 0.f, 0.f);
    for (; i < n4; i += stride) ws[i] = z;
}

// ---------------------------------------------------------------------------
// COO SpMM: dst[row,:] += val * src[col,:], D=128.
// One wave per nonzero: 32 lanes x float4 = coalesced 512B row gather,
// 4 global_atomic_add_f32 per lane for the scatter.
// ---------------------------------------------------------------------------
__global__ void spmm_atomic_kernel(const int*   __restrict__ rows,
                                   const int*   __restrict__ cols,
                                   const float* __restrict__ vals,
                                   const float* __restrict__ src,
                                   float*       __restrict__ dst,
                                   int nnz) {
    int wave = (int)((blockIdx.x * (long)blockDim.x + threadIdx.x) >> 5);
    int lane = threadIdx.x & 31;
    if (wave >= nnz) return;
    int   r = rows[wave];
    int   c = cols[wave];
    float v = vals[wave];
    const float4* s = (const float4*)(src + (size_t)c * DIM);
    float4 x = s[lane];
    float* d = dst + (size_t)r * DIM + lane * 4;
    atomicAdd(d + 0, v * x.x);
    atomicAdd(d + 1, v * x.y);
    atomicAdd(d + 2, v * x.z);
    atomicAdd(d + 3, v * x.w);
}

// ---------------------------------------------------------------------------
// Fused bi-interaction epilogue with fp32 WMMA:
//   out = leaky((ego+side) @ W1^T + b1) + leaky((ego*side) @ W2^T + b2)
// One wave per 16x16 output tile; both GEMM branches computed together so the
// ego/side A-operand loads are shared. K=128 -> 32 x V_WMMA_F32_16X16X4_F32
// per branch. Grid is exact (50000 waves), so EXEC is all ones for WMMA.
// ---------------------------------------------------------------------------
__global__ void fused_bi_wmma_kernel(const float* __restrict__ ego,
                                     const float* __restrict__ side,
                                     const float* __restrict__ W1,
                                     const float* __restrict__ b1,
                                     const float* __restrict__ W2,
                                     const float* __restrict__ b2,
                                     float*       __restrict__ out) {
    int wave  = (int)((blockIdx.x * (long)blockDim.x + threadIdx.x) >> 5);
    int lane  = threadIdx.x & 31;
    int ntile = wave & 7;                 // 8 tiles of 16 columns
    int mtile = wave >> 3;                // 6250 tiles of 16 rows
    if (mtile >= N_ENT / 16) return;      // never taken with exact grid

    int half = lane >> 4;                 // 0: K pair {0,1}, 1: K pair {2,3}
    int l16  = lane & 15;

    // A operand: row = mtile*16 + l16, k = k0 + 2*half + {0,1}  (float2 load)
    const float* egoRow  = ego  + (size_t)(mtile * 16 + l16) * DIM;
    const float* sideRow = side + (size_t)(mtile * 16 + l16) * DIM;
    // B operand: B[k][n] = W[n][k]; n = ntile*16 + l16, contiguous k pair
    const float* w1Row = W1 + (size_t)(ntile * 16 + l16) * DIM;
    const float* w2Row = W2 + (size_t)(ntile * 16 + l16) * DIM;

    v8f acc1 = {0.f, 0.f, 0.f, 0.f, 0.f, 0.f, 0.f, 0.f};
    v8f acc2 = {0.f, 0.f, 0.f, 0.f, 0.f, 0.f, 0.f, 0.f};

    #pragma unroll 4
    for (int k0 = 0; k0 < DIM; k0 += 4) {
        int k = k0 + half * 2;
        float2 e = *(const float2*)(egoRow  + k);
        float2 s = *(const float2*)(sideRow + k);
        v2f aS = { e.x + s.x, e.y + s.y };   // (ego + side) for W1 branch
        v2f aM = { e.x * s.x, e.y * s.y };   // (ego * side) for W2 branch
        float2 w1 = *(const float2*)(w1Row + k);
        float2 w2 = *(const float2*)(w2Row + k);
        v2f bW1 = { w1.x, w1.y };
        v2f bW2 = { w2.x, w2.y };
        acc1 = __builtin_amdgcn_wmma_f32_16x16x4_f32(
                   false, aS, false, bW1, (short)0, acc1, false, false);
        acc2 = __builtin_amdgcn_wmma_f32_16x16x4_f32(
                   false, aM, false, bW2, (short)0, acc2, false, false);
    }

    int   ncol  = ntile * 16 + l16;
    float bias1 = b1[ncol];
    float bias2 = b2[ncol];

    #pragma unroll
    for (int r = 0; r < 8; ++r) {
        float v1 = acc1[r] + bias1;
        float v2 = acc2[r] + bias2;
        v1 = (v1 > 0.f) ? v1 : 0.01f * v1;   // LeakyReLU(0.01)
        v2 = (v2 > 0.f) ? v2 : 0.01f * v2;
        int row = mtile * 16 + half * 8 + r; // C/D layout: lanes 16-31 -> M+8
        out[(size_t)row * DIM + ncol] = v1 + v2;
    }
}

// ---------------------------------------------------------------------------
// Host-side orchestration (graph-capture safe: only kernel launches on stream)
// ---------------------------------------------------------------------------
extern "C" void kernel_launch(void* const* d_in, const int* in_sizes, int n_in,
                              void* d_out, int out_size, void* d_ws, size_t ws_size,
                              hipStream_t stream) {
    const float* ego     = (const float*)d_in[0];
    const int*   A_rows  = (const int*)  d_in[1];
    const int*   A_cols  = (const int*)  d_in[2];
    const float* A_vals  = (const float*)d_in[3];
    const int*   p1_rows = (const int*)  d_in[4];
    const int*   p1_cols = (const int*)  d_in[5];
    const float* p1_vals = (const float*)d_in[6];
    const int*   p2_rows = (const int*)  d_in[7];
    const int*   p2_cols = (const int*)  d_in[8];
    const float* p2_vals = (const float*)d_in[9];
    const int*   l1_rows = (const int*)  d_in[10];
    const int*   l1_cols = (const int*)  d_in[11];
    const float* l1_vals = (const float*)d_in[12];
    const int*   l2_rows = (const int*)  d_in[13];
    const int*   l2_cols = (const int*)  d_in[14];
    const float* l2_vals = (const float*)d_in[15];
    const float* W1      = (const float*)d_in[16];
    const float* b1      = (const float*)d_in[17];
    const float* W2      = (const float*)d_in[18];
    const float* b2      = (const float*)d_in[19];
    float*       out     = (float*)d_out;

    // Workspace layout: side (N_ENT x D) | he1 (N_HE x D) | he2 (N_HE x D)
    float* side = (float*)d_ws;
    float* he1  = side + (size_t)N_ENT * DIM;
    float* he2  = he1  + (size_t)N_HE  * DIM;

    // 1) zero accumulators (ws is poisoned by the harness)
    long n4 = ((long)N_ENT * DIM + 2L * N_HE * DIM) / 4;
    zero_ws_kernel<<<2048, 256, 0, stream>>>((float4*)d_ws, n4);

    // 2) side = A @ ego
    int nnzA = in_sizes[1];
    spmm_atomic_kernel<<<(nnzA + 7) / 8, 256, 0, stream>>>(
        A_rows, A_cols, A_vals, ego, side, nnzA);

    // 3) he1 = p1 @ ego ; he2 = l1 @ ego
    int nnzP1 = in_sizes[4];
    spmm_atomic_kernel<<<(nnzP1 + 7) / 8, 256, 0, stream>>>(
        p1_rows, p1_cols, p1_vals, ego, he1, nnzP1);
    int nnzL1 = in_sizes[10];
    spmm_atomic_kernel<<<(nnzL1 + 7) / 8, 256, 0, stream>>>(
        l1_rows, l1_cols, l1_vals, ego, he2, nnzL1);

    // 4) side += p2 @ he1 ; side += l2 @ he2
    int nnzP2 = in_sizes[7];
    spmm_atomic_kernel<<<(nnzP2 + 7) / 8, 256, 0, stream>>>(
        p2_rows, p2_cols, p2_vals, he1, side, nnzP2);
    int nnzL2 = in_sizes[13];
    spmm_atomic_kernel<<<(nnzL2 + 7) / 8, 256, 0, stream>>>(
        l2_rows, l2_cols, l2_vals, he2, side, nnzL2);

    // 5) fused bi-interaction GEMMs + bias + LeakyReLU + add (fp32 WMMA)
    //    50000 waves = 6250 M-tiles * 8 N-tiles; 8 waves per 256-thread block
    int nblocks = (N_ENT / 16) * 8 * 32 / 256;   // 6250
    fused_bi_wmma_kernel<<<nblocks, 256, 0, stream>>>(
        ego, side, W1, b1, W2, b2, out);
}